// HGCFModel_17317308137941
// MI455X (gfx1250) — compile-verified
//
#include <hip/hip_runtime.h>
#include <hip/hip_bf16.h>

typedef __attribute__((ext_vector_type(2))) float v2f;
typedef __attribute__((ext_vector_type(8))) float v8f;

// Native f32 FP atomic (global_atomic_add_f32) — avoid CAS-loop lowering that
// plain atomicAdd(float*) can produce without unsafe-fp-atomics flags.
__device__ inline void atomic_add_f32(float* p, float v) {
#if defined(__HIP_DEVICE_COMPILE__)
    unsafeAtomicAdd(p, v);
#else
    atomicAdd(p, v);
#endif
}

// ---------------- zero workspace + output ----------------
__global__ void k_zero(float* __restrict__ h1, float* __restrict__ h2,
                       float* __restrict__ out, long n) {
    long i = (long)blockIdx.x * blockDim.x + threadIdx.x;
    long stride = (long)gridDim.x * blockDim.x;
    for (long j = i; j < n; j += stride) { h1[j] = 0.0f; h2[j] = 0.0f; }
    if (i == 0) out[0] = 0.0f;
}

// ---------------- x_t = logmap0(proj(expmap0(proj_tan0(emb)))) ----------------
// One wave32 per row; each lane owns dims {2*lane, 2*lane+1}. K = 1.
__global__ void k_tangent(const float* __restrict__ emb, float* __restrict__ xt, int nrows) {
    int idx  = blockIdx.x * blockDim.x + threadIdx.x;
    int row  = idx >> 5;
    int lane = idx & 31;
    if (row >= nrows) return;
    const float* r = emb + (size_t)row * 64;
    float e0 = r[lane * 2];
    float e1 = r[lane * 2 + 1];
    float m0 = (lane == 0) ? 0.0f : e0;              // proj_tan0 zeroes dim 0
    float s  = m0 * m0 + e1 * e1;
#pragma unroll
    for (int off = 16; off; off >>= 1) s += __shfl_xor(s, off, 32);
    // expmap0: xn = ||x||, res = [cosh, sinh*x/xn]
    float xn  = fmaxf(sqrtf(s), 1e-15f);
    float sc  = sinhf(xn) / xn;
    float yn2 = sc * sc * s;                          // ||y||^2 of hyperboloid tail
    float t   = sqrtf(fmaxf(1.0f + yn2, 1e-7f));      // proj time component
    // logmap0: rest = acosh(max(t,1+eps)) * y / ||y||
    float yn  = fmaxf(sqrtf(yn2), 1e-15f);
    float th  = fmaxf(t, 1.0f + 1e-7f);
    float f   = acoshf(th) * sc / yn;
    float* o = xt + (size_t)row * 64;
    o[lane * 2]     = (lane == 0) ? 0.0f : f * e0;    // logmap0 zeroes dim 0
    o[lane * 2 + 1] = f * e1;
}

// ---------------- out[row] += w * x[col]  (segment_sum) ----------------
// One wave32 per edge; lane gathers a float2 -> 256B coalesced row per wave.
// x fits in L2 (38.4MB < 192MB): gathers and native f32 atomics resolve in L2.
__global__ void k_spmm(const int* __restrict__ ei, const float* __restrict__ ew,
                       const float* __restrict__ x, float* __restrict__ out, int ecount) {
    int idx  = blockIdx.x * blockDim.x + threadIdx.x;
    int e    = idx >> 5;
    int lane = idx & 31;
    if (e >= ecount) return;
    int   row = ei[e];
    int   col = ei[ecount + e];
    float w   = ew[e];
    v2f v = *(const v2f*)(x + (size_t)col * 64 + lane * 2);
    float* o = out + (size_t)row * 64 + lane * 2;
    atomic_add_f32(o,     w * v.x);
    atomic_add_f32(o + 1, w * v.y);
}

// ---------------- h = proj(expmap0(h1 + h2)) ----------------
__global__ void k_hyper(const float* __restrict__ h1, const float* __restrict__ h2,
                        float* __restrict__ h, int nrows) {
    int idx  = blockIdx.x * blockDim.x + threadIdx.x;
    int row  = idx >> 5;
    int lane = idx & 31;
    if (row >= nrows) return;
    size_t base = (size_t)row * 64 + lane * 2;
    float a0 = h1[base] + h2[base];
    float a1 = h1[base + 1] + h2[base + 1];
    float m0 = (lane == 0) ? 0.0f : a0;               // expmap0 ignores dim 0
    float s  = m0 * m0 + a1 * a1;
#pragma unroll
    for (int off = 16; off; off >>= 1) s += __shfl_xor(s, off, 32);
    float xn  = fmaxf(sqrtf(s), 1e-15f);
    float sc  = sinhf(xn) / xn;
    float yn2 = sc * sc * s;
    float t   = sqrtf(fmaxf(1.0f + yn2, 1e-7f));
    h[base]     = (lane == 0) ? t : sc * a0;
    h[base + 1] = sc * a1;
}

// select c[r] with a cndmask tree (avoid scratch spill from dynamic indexing)
__device__ inline float pick8(v8f c, int r) {
    float v01 = (r & 1) ? c[1] : c[0];
    float v23 = (r & 1) ? c[3] : c[2];
    float v45 = (r & 1) ? c[5] : c[4];
    float v67 = (r & 1) ? c[7] : c[6];
    float lo  = (r & 2) ? v23 : v01;
    float hi  = (r & 2) ? v67 : v45;
    return (r & 4) ? hi : lo;
}

// ---------------- hinge loss over triples via f32 WMMA ----------------
// 16 triples per wave. Cross-dot tiles u*pos^T and u*neg^T via 16 chained
// V_WMMA_F32_16X16X4_F32 (K=64); diagonals are the needed dot products.
// A(16x4) lane layout: lane L holds A[L&15][4k+2*(L>>4) + {0,1}]  -> float2
// B(4x16) lane layout: lane L holds B[4k+2*(L>>4)+{0,1}][L&15]    -> float2 of item row
// C diag m: lane m reg m (m<8) or lane m+16 reg m-8 (m>=8).
__global__ void k_loss(const float* __restrict__ h, const int* __restrict__ triples,
                       float* __restrict__ out, int ntiles) {
    int idx  = blockIdx.x * blockDim.x + threadIdx.x;
    int wave = idx >> 5;
    int lane = idx & 31;
    if (wave >= ntiles) return;                       // whole-wave uniform exit
    int m    = lane & 15;
    int half = lane >> 4;
    int t    = wave * 16 + m;
    int uidx = triples[3 * t];
    int pidx = triples[3 * t + 1];
    int nidx = triples[3 * t + 2];
    const float* up = h + (size_t)uidx * 64;
    const float* pp = h + (size_t)pidx * 64;
    const float* nq = h + (size_t)nidx * 64;

    v8f cpos = {0.f, 0.f, 0.f, 0.f, 0.f, 0.f, 0.f, 0.f};
    v8f cneg = {0.f, 0.f, 0.f, 0.f, 0.f, 0.f, 0.f, 0.f};
    int off = 2 * half;
#pragma unroll
    for (int k = 0; k < 16; ++k) {
        v2f a  = *(const v2f*)(up + off);
        v2f bp = *(const v2f*)(pp + off);
        v2f bn = *(const v2f*)(nq + off);
        cpos = __builtin_amdgcn_wmma_f32_16x16x4_f32(false, a, false, bp,
                                                     (short)0, cpos, false, false);
        cneg = __builtin_amdgcn_wmma_f32_16x16x4_f32(false, a, false, bn,
                                                     (short)0, cneg, false, false);
        off += 4;
    }

    bool active = (lane < 8) || (lane >= 24);         // lanes owning a diagonal
    float loss = 0.0f;
    if (active) {
        int reg  = lane & 7;
        float dp = pick8(cpos, reg);                  // dot(u_m, pos_m)
        float dn = pick8(cneg, reg);                  // dot(u_m, neg_m)
        float u0 = up[0], p0 = pp[0], n0 = nq[0];
        float mp = dp - 2.0f * u0 * p0;               // minkowski_dot
        float mn = dn - 2.0f * u0 * n0;
        float thp = fmaxf(-mp, 1.0f + 1e-7f);
        float thn = fmaxf(-mn, 1.0f + 1e-7f);
        float ap = acoshf(thp), an = acoshf(thn);
        float sp = fminf(ap * ap, 50.0f);             // sqdist clamp
        float sn = fminf(an * an, 50.0f);
        loss = fmaxf(sp - sn + 0.1f, 0.0f);           // MARGIN = 0.1
    }
#pragma unroll
    for (int o = 16; o; o >>= 1) loss += __shfl_xor(loss, o, 32);
    if (lane == 0) atomic_add_f32(out, loss);
}

extern "C" void kernel_launch(void* const* d_in, const int* in_sizes, int n_in,
                              void* d_out, int out_size, void* d_ws, size_t ws_size,
                              hipStream_t stream) {
    const float* emb         = (const float*)d_in[0];
    const int*   edge_index  = (const int*)d_in[1];
    const float* edge_weight = (const float*)d_in[2];
    const int*   triples     = (const int*)d_in[3];
    float* out = (float*)d_out;

    const int  nrows = in_sizes[0] / 64;   // 150000
    const int  E     = in_sizes[2];        // 1200000
    const int  B     = in_sizes[3] / 3;    // 200000
    const long N64   = (long)nrows * 64;

    float* xt = (float*)d_ws;              // N*64
    float* h1 = xt + N64;                  // N*64
    float* h2 = h1 + N64;                  // N*64
    float* h  = xt;                        // reuse xt after spmm passes

    k_zero<<<1024, 256, 0, stream>>>(h1, h2, out, N64);

    int blk = 256;
    long rowThreads = (long)nrows * 32;
    k_tangent<<<(int)((rowThreads + blk - 1) / blk), blk, 0, stream>>>(emb, xt, nrows);

    long edgeThreads = (long)E * 32;
    k_spmm<<<(int)((edgeThreads + blk - 1) / blk), blk, 0, stream>>>(edge_index, edge_weight, xt, h1, E);
    k_spmm<<<(int)((edgeThreads + blk - 1) / blk), blk, 0, stream>>>(edge_index, edge_weight, h1, h2, E);

    k_hyper<<<(int)((rowThreads + blk - 1) / blk), blk, 0, stream>>>(h1, h2, h, nrows);

    int ntiles = B / 16;                   // B divisible by 16 -> no partial tiles
    long lossThreads = (long)ntiles * 32;
    k_loss<<<(int)((lossThreads + blk - 1) / blk), blk, 0, stream>>>(h, triples, out, ntiles);
}